// TopKMaskHW_36902359007388
// MI455X (gfx1250) — compile-verified
//
#include <hip/hip_runtime.h>
#include <stdint.h>

// Problem constants (reference fixes n=32, c=256, h=w=64).
#define HW      4096                 // elements per (n,c) row
#define THREADS 256                  // 8 wave32 per block
#define ELEMS   (HW / THREADS)       // 16 elements per thread
#define NBINS   2048                 // 11-bit radix digit
#define BPT     (NBINS / THREADS)    // 8 bins owned per thread

#define AS1 __attribute__((address_space(1)))
#define AS3 __attribute__((address_space(3)))

typedef int v4i __attribute__((ext_vector_type(4)));

#if __has_builtin(__builtin_amdgcn_global_load_async_to_lds_b128)
#define HAVE_ASYNC_COPY 1
#endif

__device__ __forceinline__ AS1 v4i* glb128(const void* p) {
  // Global flat addresses are identity-mapped to addrspace(1).
  return (AS1 v4i*)(uintptr_t)p;
}
__device__ __forceinline__ AS3 v4i* lds128(void* p) {
  // Flat LDS address: low 32 bits are the LDS offset (ISA 10.2 aperture rules).
  return (AS3 v4i*)(uint32_t)(uintptr_t)p;
}

__device__ __forceinline__ void wait_async_zero() {
#if __has_builtin(__builtin_amdgcn_s_wait_asynccnt)
  __builtin_amdgcn_s_wait_asynccnt(0);
#else
  asm volatile("s_wait_asynccnt 0" ::: "memory");
#endif
}

__global__ __launch_bounds__(THREADS)
void TopKMaskHW_36902359007388_kernel(const float* __restrict__ x,
                                      const float* __restrict__ tau_p,
                                      const int* __restrict__ k_p,
                                      float* __restrict__ out) {
  __shared__ float    s_x[HW];        // 16 KB staged row
  __shared__ unsigned s_hist[NBINS];  // 8 KB radix histogram
  __shared__ unsigned s_scan[THREADS];
  __shared__ unsigned s_sel[2];       // {new prefix, new k}

  const unsigned tid      = threadIdx.x;
  const size_t   row_base = (size_t)blockIdx.x * HW;
  const float*   rowp     = x + row_base;

  // ---- Stage the row into LDS via gfx1250 async global->LDS DMA ----
#if defined(HAVE_ASYNC_COPY)
  #pragma unroll
  for (int i = 0; i < 4; ++i) {
    const unsigned off = (i * THREADS + tid) * 4;  // 16B granules, coalesced
    __builtin_amdgcn_global_load_async_to_lds_b128(
        glb128(rowp + off), lds128(&s_x[off]), /*offset=*/0, /*cpol=*/0);
  }
  wait_async_zero();   // each wave drains its own ASYNCcnt
  __syncthreads();     // then every wave's LDS writes are visible
#else
  #pragma unroll
  for (int i = 0; i < 4; ++i) {
    const unsigned off = (i * THREADS + tid) * 4;
    *(float4*)&s_x[off] = *(const float4*)(rowp + off);
  }
  __syncthreads();
#endif

  // ---- Cache this thread's strided slice (bank-conflict-free b32 reads) ----
  unsigned key[ELEMS];
  float    val[ELEMS];
  #pragma unroll
  for (int j = 0; j < ELEMS; ++j) {
    const float v = s_x[j * THREADS + tid];
    val[j] = v;
    key[j] = __float_as_uint(v) & 0x7fffffffu;  // |x| bits: u32-order == f32-order
  }

  const float tau  = tau_p[0];
  const int   keep = k_p[0];

  if (keep >= HW) {  // keep-everything fast path: out = x
    #pragma unroll
    for (int j = 0; j < ELEMS; ++j)
      out[row_base + j * THREADS + tid] = val[j];
    return;
  }

  unsigned kk     = (keep < 1) ? 1u : (unsigned)keep;
  unsigned prefix = 0u;

  // ---- 31-bit radix select: digits of 11 / 11 / 9 bits ----
  const int shifts[3] = {20, 9, 0};
  const int widths[3] = {11, 11, 9};

  for (int r = 0; r < 3; ++r) {
    const int      s          = shifts[r];
    const int      w          = widths[r];
    const unsigned nb         = 1u << w;
    const unsigned above_mask = ~((1u << (s + w)) - 1u);  // bits above this digit

    for (unsigned b = tid; b < NBINS; b += THREADS) s_hist[b] = 0u;
    __syncthreads();

    #pragma unroll
    for (int j = 0; j < ELEMS; ++j) {
      const unsigned kj = key[j];
      if ((kj & above_mask) == prefix)
        atomicAdd(&s_hist[(kj >> s) & (nb - 1u)], 1u);
    }
    __syncthreads();

    // Per-thread partials over its 8 bins.
    unsigned cnt[BPT];
    unsigned local = 0u;
    const unsigned base = tid * BPT;
    #pragma unroll
    for (int b = 0; b < BPT; ++b) { cnt[b] = s_hist[base + b]; local += cnt[b]; }

    // Inclusive suffix scan over the 256 thread partials (higher bin = larger key).
    s_scan[tid] = local;
    __syncthreads();
    unsigned acc = local;
    for (int off = 1; off < THREADS; off <<= 1) {
      const unsigned v = (tid + off < THREADS) ? s_scan[tid + off] : 0u;
      __syncthreads();
      acc += v;
      s_scan[tid] = acc;
      __syncthreads();
    }
    unsigned above = acc - local;  // # keys in bins strictly above my bin range

    // Unique bin where the descending cumulative count crosses kk.
    for (int b = BPT - 1; b >= 0; --b) {
      const unsigned c = cnt[b];
      if (above < kk && above + c >= kk) {
        s_sel[0] = prefix | ((base + (unsigned)b) << s);
        s_sel[1] = kk - above;
      }
      above += c;
    }
    __syncthreads();
    prefix = s_sel[0];
    kk     = s_sel[1];
    __syncthreads();
  }

  // ---- Blend and write (coalesced). Mirror reference arithmetic exactly. ----
  const unsigned T   = prefix;        // exact bit pattern of the k-th largest |x|
  const float    omt = 1.0f - tau;
  #pragma unroll
  for (int j = 0; j < ELEMS; ++j) {
    const float v  = val[j];
    const float sp = (key[j] >= T) ? v : 0.0f;   // sparse_x = mask * x
    out[row_base + j * THREADS + tid] =
        __fadd_rn(__fmul_rn(sp, tau), __fmul_rn(v, omt));
  }
}

extern "C" void kernel_launch(void* const* d_in, const int* in_sizes, int n_in,
                              void* d_out, int out_size, void* d_ws, size_t ws_size,
                              hipStream_t stream) {
  const float* x     = (const float*)d_in[0];
  const float* tau   = (const float*)d_in[1];
  const int*   keep  = (const int*)d_in[2];
  float*       out   = (float*)d_out;

  const int rows = in_sizes[0] / HW;  // 32*256 = 8192 rows of 4096
  TopKMaskHW_36902359007388_kernel<<<dim3(rows), dim3(THREADS), 0, stream>>>(
      x, tau, keep, out);
}